// GLQuasiAlpha_Pooled_All_Improved_52055003627862
// MI455X (gfx1250) — compile-verified
//
#include <hip/hip_runtime.h>
#include <hip/hip_bf16.h>

typedef __attribute__((ext_vector_type(2))) float v2f;
typedef __attribute__((ext_vector_type(8))) float v8f;

#define LM1 3
#define BATCH 64
#define DDIM 512
#define HDIM 64
#define FIN 16
#define NW (DDIM * DDIM)   // 262144 elements per w row
#define NROWS (LM1 * BATCH)

// ---- monotone float<->uint key mapping (total order, matches IEEE compare) ----
__device__ __forceinline__ unsigned f2k(float x) {
    unsigned u = __float_as_uint(x);
    return (u & 0x80000000u) ? ~u : (u | 0x80000000u);
}
__device__ __forceinline__ float k2f(unsigned k) {
    unsigned u = (k & 0x80000000u) ? (k & 0x7FFFFFFFu) : ~k;
    return __uint_as_float(u);
}

// =====================================================================
// Kernel 1: per-row robust stats of ws via multi-rank radix select.
// grid = 192 blocks, block = 1024 threads; one block owns one row.
// Ranks needed (n = 262144): 65535,65536 (q.25), 131071 (=median),
// 131072 (q.5 hi), 196607,196608 (q.75); min/max by direct reduce.
// =====================================================================
__global__ __launch_bounds__(1024) void wstats_kernel(const float* __restrict__ ws,
                                                      float* __restrict__ wfeat,
                                                      float* __restrict__ gmean) {
    const int row = blockIdx.x;
    const int tid = threadIdx.x;
    const float* base = ws + (size_t)row * NW;

    __shared__ unsigned hist[6][256];
    __shared__ unsigned sprefix[6];
    __shared__ int      srem[6];
    __shared__ float    sval[6];
    __shared__ float    redf[1024];
    __shared__ unsigned redu[1024];
    __shared__ float    s_sum, s_min, s_max;

    // ---- Pass A: sum + min/max keys ----
    float lsum = 0.0f;
    unsigned lmin = 0xFFFFFFFFu, lmax = 0u;
    for (int i = tid; i < NW; i += 1024) {
        float x = base[i];
        lsum += x;
        unsigned k = f2k(x);
        lmin = lmin < k ? lmin : k;
        lmax = lmax > k ? lmax : k;
    }
    redf[tid] = lsum;
    __syncthreads();
    for (int s = 512; s > 0; s >>= 1) { if (tid < s) redf[tid] += redf[tid + s]; __syncthreads(); }
    if (tid == 0) s_sum = redf[0];
    redu[tid] = lmin;
    __syncthreads();
    for (int s = 512; s > 0; s >>= 1) { if (tid < s) redu[tid] = redu[tid] < redu[tid+s] ? redu[tid] : redu[tid+s]; __syncthreads(); }
    if (tid == 0) s_min = k2f(redu[0]);
    __syncthreads();
    redu[tid] = lmax;
    __syncthreads();
    for (int s = 512; s > 0; s >>= 1) { if (tid < s) redu[tid] = redu[tid] > redu[tid+s] ? redu[tid] : redu[tid+s]; __syncthreads(); }
    if (tid == 0) s_max = k2f(redu[0]);

    // ---- Multi-rank radix select: 4 rounds of 8 bits ----
    if (tid < 6) {
        const int R[6] = {65535, 65536, 131071, 131072, 196607, 196608};
        sprefix[tid] = 0u;
        srem[tid]    = R[tid];
    }
    __syncthreads();

    for (int round = 0; round < 4; ++round) {
        const int shift = 24 - 8 * round;
        for (int i = tid; i < 6 * 256; i += 1024) ((unsigned*)hist)[i] = 0u;
        __syncthreads();
        unsigned pref[6];
        if (round > 0) {
            #pragma unroll
            for (int r = 0; r < 6; ++r) pref[r] = sprefix[r];
        }
        for (int i = tid; i < NW; i += 1024) {
            unsigned k = f2k(base[i]);
            unsigned byte = (k >> shift) & 255u;
            if (round == 0) {
                atomicAdd(&hist[0][byte], 1u);
            } else {
                unsigned hi = k >> (shift + 8);
                #pragma unroll
                for (int r = 0; r < 6; ++r)
                    if (hi == pref[r]) atomicAdd(&hist[r][byte], 1u);
            }
        }
        __syncthreads();
        if (tid < 6) {
            const int r = tid;
            const unsigned* h = (round == 0) ? &hist[0][0] : &hist[r][0];
            int rem = srem[r];
            int cum = 0;
            for (int b = 0; b < 256; ++b) {
                int c = (int)h[b];
                if (rem < cum + c) {
                    srem[r] = rem - cum;
                    sprefix[r] = (sprefix[r] << 8) | (unsigned)b;
                    break;
                }
                cum += c;
            }
        }
        __syncthreads();
    }
    if (tid < 6) sval[tid] = k2f(sprefix[tid]);
    __syncthreads();
    const float med = sval[2];

    // ---- MAD: rank 131071 of |x - med| (non-negative floats: bits monotone) ----
    if (tid == 0) { sprefix[0] = 0u; srem[0] = 131071; }
    __syncthreads();
    for (int round = 0; round < 4; ++round) {
        const int shift = 24 - 8 * round;
        for (int i = tid; i < 256; i += 1024) hist[0][i] = 0u;
        __syncthreads();
        const unsigned pref = sprefix[0];
        for (int i = tid; i < NW; i += 1024) {
            unsigned k = __float_as_uint(fabsf(base[i] - med));
            unsigned hi = (round == 0) ? 0u : (k >> (shift + 8));
            if (round == 0 || hi == pref)
                atomicAdd(&hist[0][(k >> shift) & 255u], 1u);
        }
        __syncthreads();
        if (tid == 0) {
            int rem = srem[0];
            int cum = 0;
            for (int b = 0; b < 256; ++b) {
                int c = (int)hist[0][b];
                if (rem < cum + c) {
                    srem[0] = rem - cum;
                    sprefix[0] = (sprefix[0] << 8) | (unsigned)b;
                    break;
                }
                cum += c;
            }
        }
        __syncthreads();
    }

    if (tid == 0) {
        float mad = __uint_as_float(sprefix[0]);
        float* o = wfeat + row * 7;
        o[0] = med;
        o[1] = mad;
        o[2] = s_min;                               // q=0.00
        o[3] = 0.25f * sval[0] + 0.75f * sval[1];   // q=0.25, idx 65535.75
        o[4] = 0.5f  * (sval[2] + sval[3]);         // q=0.50, idx 131071.5
        o[5] = 0.75f * sval[4] + 0.25f * sval[5];   // q=0.75, idx 196607.25
        o[6] = s_max;                               // q=1.00
        gmean[row] = s_sum * (1.0f / (float)NW);
    }
}

// =====================================================================
// Kernel 2: per-row robust stats of bs (512 elems) via rank counting.
// grid = 192 blocks, block = 512 threads.
// =====================================================================
__global__ __launch_bounds__(512) void bstats_kernel(const float* __restrict__ bs,
                                                     float* __restrict__ bfeat) {
    const int row = blockIdx.x;
    const int t = threadIdx.x;
    __shared__ float sv[512];
    __shared__ float slot[8];
    __shared__ float madv;

    const float v = bs[row * 512 + t];
    sv[t] = v;
    __syncthreads();
    int rank = 0;
    for (int j = 0; j < 512; ++j) {
        float u = sv[j];
        rank += (u < v) || (u == v && j < t);
    }
    if      (rank == 0)   slot[0] = v;
    else if (rank == 127) slot[1] = v;
    else if (rank == 128) slot[2] = v;
    else if (rank == 255) slot[3] = v;
    else if (rank == 256) slot[4] = v;
    else if (rank == 383) slot[5] = v;
    else if (rank == 384) slot[6] = v;
    else if (rank == 511) slot[7] = v;
    __syncthreads();
    const float med = slot[3];                 // lower median, idx (512-1)//2 = 255
    const float d = fabsf(v - med);
    sv[t] = d;
    __syncthreads();
    int rank2 = 0;
    for (int j = 0; j < 512; ++j) {
        float u = sv[j];
        rank2 += (u < d) || (u == d && j < t);
    }
    if (rank2 == 255) madv = d;
    __syncthreads();
    if (t == 0) {
        float* o = bfeat + row * 7;
        o[0] = med;
        o[1] = madv;
        o[2] = slot[0];                             // q=0.00
        o[3] = 0.25f * slot[1] + 0.75f * slot[2];   // q=0.25, idx 127.75
        o[4] = 0.5f  * (slot[3] + slot[4]);         // q=0.50, idx 255.5
        o[5] = 0.75f * slot[5] + 0.25f * slot[6];   // q=0.75, idx 383.25
        o[6] = slot[7];                             // q=1.00
    }
}

// =====================================================================
// Kernel 3: features -> LN -> fc1 -> LN(aff) -> GELU(erf) -> gate ->
//           fco (WMMA f32 16x16x4) -> sin*scale + 1.
// grid = 3 blocks (one per layer l), block = 256 threads (8 waves).
// =====================================================================
__global__ __launch_bounds__(256) void mlp_kernel(const float* __restrict__ wfeat,
                                                  const float* __restrict__ bfeat,
                                                  const float* __restrict__ gmean,
                                                  const float* __restrict__ fc1_w,
                                                  const float* __restrict__ fc1_b,
                                                  const float* __restrict__ ln_w,
                                                  const float* __restrict__ ln_b,
                                                  const float* __restrict__ gate_w,
                                                  const float* __restrict__ gate_b,
                                                  const float* __restrict__ fco_w,
                                                  const float* __restrict__ fco_b,
                                                  const float* __restrict__ scale,
                                                  float* __restrict__ out) {
    const int l = blockIdx.x;
    const int tid = threadIdx.x;

    __shared__ float xs[BATCH][FIN];
    __shared__ float hs[BATCH][HDIM + 1];
    __shared__ float hg[BATCH][HDIM + 1];
    __shared__ float mu_s[BATCH], rs_s[BATCH];

    // --- combined features + LayerNorm (g_mad feature == 0 exactly) ---
    if (tid < BATCH) {
        const int row = l * BATCH + tid;
        float c[16];
        #pragma unroll
        for (int j = 0; j < 7; ++j) c[j]     = wfeat[row * 7 + j];
        #pragma unroll
        for (int j = 0; j < 7; ++j) c[7 + j] = bfeat[row * 7 + j];
        c[14] = gmean[row];
        c[15] = 0.0f;
        float mu = 0.0f;
        #pragma unroll
        for (int j = 0; j < 16; ++j) mu += c[j];
        mu *= (1.0f / 16.0f);
        float var = 0.0f;
        #pragma unroll
        for (int j = 0; j < 16; ++j) { float d = c[j] - mu; var += d * d; }
        var *= (1.0f / 16.0f);
        float rs = rsqrtf(var + 1e-5f);
        #pragma unroll
        for (int j = 0; j < 16; ++j) xs[tid][j] = (c[j] - mu) * rs;
    }
    __syncthreads();

    // --- fc1: h[b][hh] = x[b] . fc1_w[l][hh] + fc1_b ---
    for (int idx = tid; idx < BATCH * HDIM; idx += 256) {
        const int b = idx >> 6, hh = idx & 63;
        const float* wrow = fc1_w + (l * HDIM + hh) * FIN;
        float acc = fc1_b[l * HDIM + hh];
        #pragma unroll
        for (int j = 0; j < FIN; ++j) acc += xs[b][j] * wrow[j];
        hs[b][hh] = acc;
    }
    __syncthreads();

    // --- LayerNorm stats over H per batch row ---
    if (tid < BATCH) {
        float mu = 0.0f;
        for (int j = 0; j < HDIM; ++j) mu += hs[tid][j];
        mu *= (1.0f / HDIM);
        float var = 0.0f;
        for (int j = 0; j < HDIM; ++j) { float d = hs[tid][j] - mu; var += d * d; }
        var *= (1.0f / HDIM);
        mu_s[tid] = mu;
        rs_s[tid] = rsqrtf(var + 1e-5f);
    }
    __syncthreads();

    // --- LN affine + exact GELU ---
    for (int idx = tid; idx < BATCH * HDIM; idx += 256) {
        const int b = idx >> 6, hh = idx & 63;
        float v = (hs[b][hh] - mu_s[b]) * rs_s[b] * ln_w[l * HDIM + hh] + ln_b[l * HDIM + hh];
        hg[b][hh] = 0.5f * v * (1.0f + erff(v * 0.7071067811865475f));
    }
    __syncthreads();

    // --- gate + multiply; result (h*gate) goes back into hs ---
    for (int idx = tid; idx < BATCH * HDIM; idx += 256) {
        const int b = idx >> 6, g = idx & 63;
        const float* wrow = gate_w + (l * HDIM + g) * HDIM;
        float acc = gate_b[l * HDIM + g];
        #pragma unroll 4
        for (int j = 0; j < HDIM; ++j) acc += hg[b][j] * wrow[j];
        float sgm = 1.0f / (1.0f + expf(-acc));
        hs[b][g] = hg[b][g] * sgm;
    }
    __syncthreads();

    // --- fco: [64x64] x [64x512] with fp32 WMMA 16x16x4, fused epilogue ---
    const int wave = tid >> 5;
    const int lane = tid & 31;
    const float* fw = fco_w + l * DDIM * HDIM;   // [o][h] row-major
    const float* fb = fco_b + l * DDIM;
    const float scl = scale[l];
    float* orow = out + (size_t)l * BATCH * DDIM;

    const int lhalf = lane >> 4;        // 0: lanes 0-15, 1: lanes 16-31
    const int l15   = lane & 15;
    const int ksel  = lhalf * 2;        // K sub-offset selected by lane half

    // 128 tiles of 16x16 (4 m-tiles x 32 n-tiles); 8 waves, uniform per-wave loop
    for (int t = wave; t < (BATCH / 16) * (DDIM / 16); t += 8) {
        const int mt = t & 3;
        const int nt = t >> 2;
        const int ml = mt * 16 + l15;   // A row for this lane
        const int nl = nt * 16 + l15;   // B col for this lane
        v8f acc = {};
        #pragma unroll 4
        for (int k0 = 0; k0 < HDIM; k0 += 4) {
            // A 16x4 f32 frag: lane=M, VGPR+half selects K (ISA layout)
            v2f a;
            a.x = hs[ml][k0 + ksel];
            a.y = hs[ml][k0 + ksel + 1];
            // B 4x16 f32 frag: lane=N, VGPR+half selects K; B[k][n] = fco_w[n][k]
            v2f bfr;
            bfr.x = fw[nl * HDIM + k0 + ksel];
            bfr.y = fw[nl * HDIM + k0 + ksel + 1];
            acc = __builtin_amdgcn_wmma_f32_16x16x4_f32(
                /*neg_a=*/false, a, /*neg_b=*/false, bfr,
                /*c_mod=*/(short)0, acc, /*reuse_a=*/false, /*reuse_b=*/false);
        }
        // D 16x16 f32 layout: VGPR j -> M = mt*16 + j + 8*lhalf, N = nt*16 + l15
        #pragma unroll
        for (int j = 0; j < 8; ++j) {
            const int m = mt * 16 + j + 8 * lhalf;
            const int n = nt * 16 + l15;
            float raw = acc[j] + fb[n];
            orow[m * DDIM + n] = sinf(raw) * scl + 1.0f;
        }
    }
}

extern "C" void kernel_launch(void* const* d_in, const int* in_sizes, int n_in,
                              void* d_out, int out_size, void* d_ws, size_t ws_size,
                              hipStream_t stream) {
    (void)in_sizes; (void)n_in; (void)out_size; (void)ws_size;
    const float* ws_p    = (const float*)d_in[0];   // [3,64,512,512]
    const float* bs_p    = (const float*)d_in[1];   // [3,64,512]
    // d_in[2] (w_last) and d_in[3] (b_last) are unused by the reference
    const float* fc1_w   = (const float*)d_in[4];
    const float* fc1_b   = (const float*)d_in[5];
    const float* ln_w    = (const float*)d_in[6];
    const float* ln_b    = (const float*)d_in[7];
    const float* gate_w  = (const float*)d_in[8];
    const float* gate_b  = (const float*)d_in[9];
    const float* fco_w   = (const float*)d_in[10];
    const float* fco_b   = (const float*)d_in[11];
    const float* scale   = (const float*)d_in[12];
    float* out = (float*)d_out;

    float* wfeat = (float*)d_ws;             // [192][7]
    float* gmean = wfeat + NROWS * 7;        // [192]
    float* bfeat = gmean + NROWS;            // [192][7]

    wstats_kernel<<<NROWS, 1024, 0, stream>>>(ws_p, wfeat, gmean);
    bstats_kernel<<<NROWS, 512, 0, stream>>>(bs_p, bfeat);
    mlp_kernel<<<LM1, 256, 0, stream>>>(wfeat, bfeat, gmean,
                                        fc1_w, fc1_b, ln_w, ln_b,
                                        gate_w, gate_b, fco_w, fco_b,
                                        scale, out);
}